// JointlyTrainModel_2259152797908
// MI455X (gfx1250) — compile-verified
//
#include <hip/hip_runtime.h>
#include <math.h>

// ---------------- problem constants ----------------
#define BGRAPHS 1024
#define NNODES  62
#define NCIN    64
#define NCOUT   128
#define NDEGE   16
#define EPG     (NNODES*NDEGE)        // 992 edges per graph
#define NTOT    (BGRAPHS*NNODES)      // 63488
#define ETOT    (NTOT*NDEGE)          // 1015808
#define HDIM    (NNODES*NCOUT)        // 7936
#define LIN1    512
#define LIN2    256
#define MPAD    64                    // padded rows per graph tile

typedef float v2f __attribute__((ext_vector_type(2)));
typedef float v8f __attribute__((ext_vector_type(8)));

// D = A(16x4 f32) x B(4x16 f32) + C(16x16 f32), wave32 WMMA
__device__ __forceinline__ v8f wmma4(v2f a, v2f b, v8f c) {
  return __builtin_amdgcn_wmma_f32_16x16x4_f32(false, a, false, b, (short)0, c,
                                               false, false);
}

// gfx1250 async global->LDS DMA (ASYNCcnt-tracked), 16B per lane.
// VDST = LDS byte address (low 32 bits of generic shared pointer),
// VADDR = 64-bit global address, GV mode (saddr = off).
__device__ __forceinline__ void async_ld_b128(unsigned lds_off,
                                              const float* g) {
  asm volatile("global_load_async_to_lds_b128 %0, %1, off"
               :: "v"(lds_off), "v"(g)
               : "memory");
}
__device__ __forceinline__ void wait_async_le4() {
  asm volatile("s_wait_asynccnt 0x4" ::: "memory");
}
__device__ __forceinline__ void wait_async_0() {
  asm volatile("s_wait_asynccnt 0x0" ::: "memory");
}

// =====================================================================
// Kernel 1: fully fused ChebConv per graph.
//   One workgroup (8 wave32s) per 62-node graph. x tile, T1, T2 live in
//   LDS; edge scatter via ds_add_f32 atomics; final 64x128 GEMM over the
//   three Chebyshev terms done with V_WMMA_F32_16X16X4_F32; bias+ReLU,
//   store h row (graph-major, node*128+c layout == reshape(B,-1)).
// =====================================================================
__global__ __launch_bounds__(256) void cheb_graph_kernel(
    const float* __restrict__ x,     // [NTOT][64]
    const int*   __restrict__ esrc,  // [ETOT]
    const int*   __restrict__ edst,  // [ETOT]
    const float* __restrict__ Wc,    // [3][64][128]
    const float* __restrict__ bc,    // [128]
    float* __restrict__ h)           // [BGRAPHS][HDIM]
{
  __shared__ float T0[MPAD][NCIN];   // 16 KB  (Tx0 = x, rows 62..63 = 0)
  __shared__ float T1[MPAD][NCIN];   // 16 KB
  __shared__ float T2[MPAD][NCIN];   // 16 KB
  __shared__ float deg[MPAD];
  __shared__ float dis[MPAD];
  __shared__ int   er[EPG];
  __shared__ int   ec[EPG];
  __shared__ float ew[EPG];          // total LDS ~60 KB

  const int g    = blockIdx.x;
  const int tid  = threadIdx.x;
  const int lane = tid & 31;
  const int wave = tid >> 5;
  const float* xg = x + (size_t)g * NNODES * NCIN;

  // ---- stage x into LDS (zero-pad rows 62/63), clear T1/T2 ----
  for (int i = tid; i < (MPAD * NCIN) / 4; i += 256) {
    int base = i * 4;
    int row  = base >> 6;
    int c    = base & 63;
    float4 v = make_float4(0.f, 0.f, 0.f, 0.f);
    if (row < NNODES) v = *(const float4*)(xg + row * NCIN + c);
    *(float4*)&T0[row][c] = v;
    *(float4*)&T1[row][c] = make_float4(0.f, 0.f, 0.f, 0.f);
    *(float4*)&T2[row][c] = make_float4(0.f, 0.f, 0.f, 0.f);
  }
  if (tid < MPAD) deg[tid] = 0.f;
  __syncthreads();

  // ---- edges (localized) + degree over src ----
  for (int e = tid; e < EPG; e += 256) {
    int r = esrc[(size_t)g * EPG + e] - g * NNODES;
    int c = edst[(size_t)g * EPG + e] - g * NNODES;
    er[e] = r;
    ec[e] = c;
    atomicAdd(&deg[r], 1.0f);
  }
  __syncthreads();
  if (tid < MPAD) dis[tid] = (deg[tid] > 0.f) ? rsqrtf(deg[tid]) : 0.f;
  __syncthreads();
  for (int e = tid; e < EPG; e += 256) ew[e] = -(dis[er[e]] * dis[ec[e]]);
  __syncthreads();

  // ---- T1 = lmul(T0): scatter-add w * x[src] into row dst ----
  for (int idx = tid; idx < EPG * (NCIN / 4); idx += 256) {
    int e = idx >> 4;
    int q = (idx & 15) << 2;
    float w = ew[e];
    int  r  = er[e], c = ec[e];
    float4 v = *(const float4*)&T0[r][q];
    atomicAdd(&T1[c][q + 0], w * v.x);
    atomicAdd(&T1[c][q + 1], w * v.y);
    atomicAdd(&T1[c][q + 2], w * v.z);
    atomicAdd(&T1[c][q + 3], w * v.w);
  }
  __syncthreads();

  // ---- T2 = 2*lmul(T1) - T0 ----
  for (int idx = tid; idx < EPG * (NCIN / 4); idx += 256) {
    int e = idx >> 4;
    int q = (idx & 15) << 2;
    float w = ew[e];
    int  r  = er[e], c = ec[e];
    float4 v = *(const float4*)&T1[r][q];
    atomicAdd(&T2[c][q + 0], w * v.x);
    atomicAdd(&T2[c][q + 1], w * v.y);
    atomicAdd(&T2[c][q + 2], w * v.z);
    atomicAdd(&T2[c][q + 3], w * v.w);
  }
  __syncthreads();
  {
    float* t2 = &T2[0][0];
    const float* t0 = &T0[0][0];
    for (int i = tid; i < MPAD * NCIN; i += 256)
      t2[i] = 2.f * t2[i] - t0[i];
  }
  __syncthreads();

  // ---- out[64][128] = sum_t T_t @ W[t], WMMA f32 16x16x4 ----
  // wave w: M-tile = w>>1 (16 rows), N-half = (w&1)*64 (4 N-tiles)
  const int mt = wave >> 1;
  const int nb = (wave & 1) * 64;
  const int m  = mt * 16 + (lane & 15);
  const int kof = (lane >> 4) << 1;     // lanes 0-15 -> K {0,1}; 16-31 -> {2,3}
  const int cl  = lane & 15;

  v8f acc[4] = {};
  const float* Tptr[3] = { &T0[0][0], &T1[0][0], &T2[0][0] };
  for (int t = 0; t < 3; ++t) {
    const float* T  = Tptr[t];
    const float* Wt = Wc + t * (NCIN * NCOUT);
#pragma unroll 4
    for (int k4 = 0; k4 < 16; ++k4) {
      int k0 = k4 * 4 + kof;
      v2f a = *(const v2f*)(T + m * NCIN + k0);      // A[m][k0], A[m][k0+1]
      const float* wp = Wt + (size_t)k0 * NCOUT + nb + cl;
#pragma unroll
      for (int nt = 0; nt < 4; ++nt) {
        v2f b;
        b.x = wp[nt * 16];                           // B[k0  ][col]
        b.y = wp[NCOUT + nt * 16];                   // B[k0+1][col]
        acc[nt] = wmma4(a, b, acc[nt]);
      }
    }
  }

  // ---- bias + ReLU + store ----
  float* hg = h + (size_t)g * HDIM;
  const int rbase = mt * 16 + ((lane >> 4) << 3);
#pragma unroll
  for (int v = 0; v < 8; ++v) {
    int r = rbase + v;
    if (r < NNODES) {
#pragma unroll
      for (int nt = 0; nt < 4; ++nt) {
        int col = nb + nt * 16 + cl;
        hg[r * NCOUT + col] = fmaxf(acc[nt][v] + bc[col], 0.f);
      }
    }
  }
}

// =====================================================================
// Kernel 2: WMMA fp32 GEMM  C[M][N] = A[M][K] @ B[K][N] + bias
//   64x64 block tile, 8 wave32s, 2 output tiles/wave.
//   K staged 32-wide with gfx1250 GLOBAL_LOAD_ASYNC_TO_LDS_B128 and
//   double-buffered LDS: chunk i+1's DMA flies while chunk i's WMMAs run.
//   Each wave issues exactly 4 async instructions per chunk; async loads
//   complete in order, so s_wait_asynccnt 4 pins the previous chunk.
// =====================================================================
__global__ __launch_bounds__(256) void gemm_bias_kernel(
    const float* __restrict__ A, const float* __restrict__ Bw,
    const float* __restrict__ bias, float* __restrict__ C,
    int M, int N, int K)
{
  __shared__ float As[2][64][32];  // 16 KB
  __shared__ float Bs[2][32][64];  // 16 KB
  const int tid  = threadIdx.x;
  const int lane = tid & 31;
  const int wave = tid >> 5;
  const int m0 = blockIdx.y * 64;
  const int n0 = blockIdx.x * 64;
  const int mt = wave >> 1;
  const int nh = wave & 1;
  const int mrow = mt * 16 + (lane & 15);
  const int kof  = (lane >> 4) << 1;
  const int cl   = lane & 15;

  // per-thread staging slots: s in {tid, tid+256} covers 512 b128 each
  const int ar0 = tid >> 3,  ak0 = (tid & 7) << 2;          // A slot 0
  const int ar1 = (tid + 256) >> 3, ak1 = ak0;              // A slot 1
  const int br0 = tid >> 4,  bc0 = (tid & 15) << 2;         // B slot 0
  const int br1 = (tid + 256) >> 4, bc1 = bc0;              // B slot 1

  auto issue = [&](int buf, int kc) {
    async_ld_b128((unsigned)(uintptr_t)&As[buf][ar0][ak0],
                  A + (size_t)(m0 + ar0) * K + kc + ak0);
    async_ld_b128((unsigned)(uintptr_t)&Bs[buf][br0][bc0],
                  Bw + (size_t)(kc + br0) * N + n0 + bc0);
    async_ld_b128((unsigned)(uintptr_t)&As[buf][ar1][ak1],
                  A + (size_t)(m0 + ar1) * K + kc + ak1);
    async_ld_b128((unsigned)(uintptr_t)&Bs[buf][br1][bc1],
                  Bw + (size_t)(kc + br1) * N + n0 + bc1);
  };

  v8f acc[2] = {};
  issue(0, 0);
  int buf = 0;
  for (int kc = 0; kc < K; kc += 32) {
    const bool more = (kc + 32) < K;
    if (more) {
      issue(buf ^ 1, kc + 32);   // prefetch next chunk (4 ops outstanding)
      wait_async_le4();          // previous chunk resident
    } else {
      wait_async_0();            // drain tail
    }
    __syncthreads();             // all waves see buffer `buf`
#pragma unroll
    for (int k4 = 0; k4 < 8; ++k4) {
      int k0 = k4 * 4 + kof;
      v2f a = *(const v2f*)&As[buf][mrow][k0];
#pragma unroll
      for (int nt = 0; nt < 2; ++nt) {
        int col = nh * 32 + nt * 16 + cl;
        v2f b;
        b.x = Bs[buf][k0][col];
        b.y = Bs[buf][k0 + 1][col];
        acc[nt] = wmma4(a, b, acc[nt]);
      }
    }
    __syncthreads();             // done reading `buf` before its refill
    buf ^= 1;
  }

  const int rbase = m0 + mt * 16 + ((lane >> 4) << 3);
#pragma unroll
  for (int v = 0; v < 8; ++v) {
#pragma unroll
    for (int nt = 0; nt < 2; ++nt) {
      int col = n0 + nh * 32 + nt * 16 + cl;
      C[(size_t)(rbase + v) * N + col] = acc[nt][v] + bias[col];
    }
  }
}

// =====================================================================
// Kernel 3: fused train-mode BatchNorm (biased var) + ReLU, in place.
//   Block = 16 columns x 16 row-threads; stats reduced in LDS.
// =====================================================================
__global__ __launch_bounds__(256) void bn_relu_kernel(
    float* __restrict__ C, const float* __restrict__ g,
    const float* __restrict__ b, int M, int N)
{
  __shared__ float s1[16][17];
  __shared__ float s2[16][17];
  __shared__ float mu[16], ri[16];
  const int tid = threadIdx.x;
  const int c   = tid & 15;
  const int r0  = tid >> 4;
  const int col = blockIdx.x * 16 + c;

  float a = 0.f, a2 = 0.f;
  for (int r = r0; r < M; r += 16) {
    float v = C[(size_t)r * N + col];
    a += v; a2 += v * v;
  }
  s1[r0][c] = a; s2[r0][c] = a2;
  __syncthreads();
  for (int off = 8; off; off >>= 1) {
    if (r0 < off) {
      s1[r0][c] += s1[r0 + off][c];
      s2[r0][c] += s2[r0 + off][c];
    }
    __syncthreads();
  }
  if (r0 == 0) {
    float mean = s1[0][c] / (float)M;
    float var  = s2[0][c] / (float)M - mean * mean;
    mu[c] = mean;
    ri[c] = rsqrtf(var + 1e-5f);
  }
  __syncthreads();
  const float gg = g[col], bb = b[col], m = mu[c], rr = ri[c];
  for (int r = r0; r < M; r += 16) {
    float v = C[(size_t)r * N + col];
    C[(size_t)r * N + col] = fmaxf((v - m) * rr * gg + bb, 0.f);
  }
}

// =====================================================================
// Kernel 4: layer-3 (K=256, N<=128) + softmax (mode 0) / L2-norm (mode 1)
//   One block per batch row; wave32 shuffle reductions.
// =====================================================================
__global__ __launch_bounds__(256) void head_final_kernel(
    const float* __restrict__ H,  // [1024][256]
    const float* __restrict__ w3, // [256][OUT]
    const float* __restrict__ b3, // [OUT]
    float* __restrict__ out,      // [1024][OUT]
    int OUT, int mode)
{
  __shared__ float part[128][8];
  __shared__ float logits[128];
  __shared__ float st[2];
  const int tid  = threadIdx.x;
  const int lane = tid & 31;
  const int wave = tid >> 5;
  const int row  = blockIdx.x;

  const float a = H[(size_t)row * LIN2 + tid];  // K == blockDim == 256
  for (int j = 0; j < OUT; ++j) {
    float p = a * w3[(size_t)tid * OUT + j];
    for (int off = 16; off; off >>= 1) p += __shfl_down(p, off, 32);
    if (lane == 0) part[j][wave] = p;
  }
  __syncthreads();
  if (tid < OUT) {
    float s = 0.f;
#pragma unroll
    for (int w = 0; w < 8; ++w) s += part[tid][w];
    logits[tid] = s + b3[tid];
  }
  __syncthreads();
  if (tid == 0) {
    if (mode == 0) {
      float mx = -1e30f;
      for (int j = 0; j < OUT; ++j) mx = fmaxf(mx, logits[j]);
      float sum = 0.f;
      for (int j = 0; j < OUT; ++j) sum += __expf(logits[j] - mx);
      st[0] = mx;
      st[1] = 1.f / sum;
    } else {
      float ss = 0.f;
      for (int j = 0; j < OUT; ++j) ss += logits[j] * logits[j];
      st[0] = 0.f;
      st[1] = 1.f / fmaxf(sqrtf(ss), 1e-12f);
    }
  }
  __syncthreads();
  if (tid < OUT) {
    float l = logits[tid];
    out[(size_t)row * OUT + tid] = (mode == 0) ? __expf(l - st[0]) * st[1]
                                               : l * st[1];
  }
}

// =====================================================================
// Launch: 5 views, pipelined through one h / c1 / c2 scratch set.
// Input layout (setup_inputs dict order, nested dicts flattened):
//   0..4  x1..x5          5..9 e1..e5 ([2][E] int32)
//   10 cheb_w  11 cheb_b
//   12.. hf{w1,b1,g1,bb1,w2,b2,g2,bb2,w3,b3}, then hs, hc, proj (10 each)
// =====================================================================
extern "C" void kernel_launch(void* const* d_in, const int* in_sizes, int n_in,
                              void* d_out, int out_size, void* d_ws, size_t ws_size,
                              hipStream_t stream) {
  (void)in_sizes; (void)n_in; (void)out_size; (void)ws_size;
  const float* xs[5];
  const int*   es[5];
  for (int i = 0; i < 5; ++i) xs[i] = (const float*)d_in[i];
  for (int i = 0; i < 5; ++i) es[i] = (const int*)d_in[5 + i];
  const float* cheb_w = (const float*)d_in[10];
  const float* cheb_b = (const float*)d_in[11];

  const int   hbase[5]  = {12, 22, 32, 42, 42};  // hf, hs, hc, proj, proj
  const int   outdim[5] = {5, 6, 3, 128, 128};
  const int   mode[5]   = {0, 0, 0, 1, 1};
  const size_t ooff[5]  = {0, 5120, 11264, 14336, 145408};

  float* h   = (float*)d_ws;                       // 1024 x 7936
  float* c1  = h  + (size_t)BGRAPHS * HDIM;        // 1024 x 512
  float* c2  = c1 + (size_t)BGRAPHS * LIN1;        // 1024 x 256
  float* outp = (float*)d_out;

  for (int v = 0; v < 5; ++v) {
    const float* w1  = (const float*)d_in[hbase[v] + 0];
    const float* b1  = (const float*)d_in[hbase[v] + 1];
    const float* g1  = (const float*)d_in[hbase[v] + 2];
    const float* bb1 = (const float*)d_in[hbase[v] + 3];
    const float* w2  = (const float*)d_in[hbase[v] + 4];
    const float* b2  = (const float*)d_in[hbase[v] + 5];
    const float* g2  = (const float*)d_in[hbase[v] + 6];
    const float* bb2 = (const float*)d_in[hbase[v] + 7];
    const float* w3  = (const float*)d_in[hbase[v] + 8];
    const float* b3  = (const float*)d_in[hbase[v] + 9];

    cheb_graph_kernel<<<BGRAPHS, 256, 0, stream>>>(
        xs[v], es[v], es[v] + ETOT, cheb_w, cheb_b, h);

    gemm_bias_kernel<<<dim3(LIN1 / 64, BGRAPHS / 64), 256, 0, stream>>>(
        h, w1, b1, c1, BGRAPHS, LIN1, HDIM);
    bn_relu_kernel<<<LIN1 / 16, 256, 0, stream>>>(c1, g1, bb1, BGRAPHS, LIN1);

    gemm_bias_kernel<<<dim3(LIN2 / 64, BGRAPHS / 64), 256, 0, stream>>>(
        c1, w2, b2, c2, BGRAPHS, LIN2, LIN1);
    bn_relu_kernel<<<LIN2 / 16, 256, 0, stream>>>(c2, g2, bb2, BGRAPHS, LIN2);

    head_final_kernel<<<BGRAPHS, 256, 0, stream>>>(
        c2, w3, b3, outp + ooff[v], outdim[v], mode[v]);
  }
}